// CNOT_layer_16140487098686
// MI455X (gfx1250) — compile-verified
//
#include <hip/hip_runtime.h>
#include <cstdint>

// Problem constants (from the reference): state is [2, 4096, 512] float32.
#define NBLOCKS   2
#define DDIM      4096
#define BATCH     512
#define ROW_BYTES (BATCH * 4)        // 2048 bytes per row
#define ROWS_TOTAL (NBLOCKS * DDIM)  // 8192 rows
#define ROWS_PER_BLOCK 8
#define CHUNKS_PER_BLOCK (ROWS_PER_BLOCK * ROW_BYTES / 16)  // 1024 x 16B chunks
#define CHUNKS_PER_THREAD 4          // 256 threads * 4 = 1024

// Composed permutation: src_row_index = p(out_row_index).
// PAIRS = [(0,1),(2,3),(5,2),(11,7)], big-endian bits (bitpos = 11 - qubit).
// out = U0 U1 U2 U3 in  =>  src = perm0(perm1(perm2(perm3(i)))): apply (11,7) first.
__device__ __forceinline__ unsigned cnot_perm(unsigned j) {
    j ^= ((j >> 0)  & 1u) << 4;   // pair (11,7): control bit 0,  target bit 4
    j ^= ((j >> 6)  & 1u) << 9;   // pair (5,2):  control bit 6,  target bit 9
    j ^= ((j >> 9)  & 1u) << 8;   // pair (2,3):  control bit 9,  target bit 8
    j ^= ((j >> 11) & 1u) << 10;  // pair (0,1):  control bit 11, target bit 10
    return j;
}

__global__ __launch_bounds__(256)
void cnot_gather_async(const float* __restrict__ in, float* __restrict__ out) {
    __shared__ __align__(16) unsigned char lds[CHUNKS_PER_BLOCK * 16];  // 16 KB

    const unsigned tid  = threadIdx.x;
    const uint64_t inB  = (uint64_t)(uintptr_t)in;
    const uint64_t outB = (uint64_t)(uintptr_t)out;
    // Flat->LDS aperture puts the wave-relative LDS byte offset in addr[31:0].
    const unsigned ldsBase = (unsigned)(uintptr_t)(&lds[0]);

    uint64_t src_addr[CHUNKS_PER_THREAD];
    uint64_t dst_addr[CHUNKS_PER_THREAD];
    unsigned lds_off[CHUNKS_PER_THREAD];

#pragma unroll
    for (int k = 0; k < CHUNKS_PER_THREAD; ++k) {
        const unsigned c         = tid + (unsigned)k * 256u;   // [0, 1024)
        const unsigned row_local = c >> 7;                     // 128 chunks per row
        const unsigned chunk     = c & 127u;
        const unsigned out_row   = blockIdx.x * ROWS_PER_BLOCK + row_local;
        const unsigned n         = out_row >> 12;
        const unsigned i         = out_row & 4095u;
        const unsigned src_row   = (n << 12) | cnot_perm(i);
        src_addr[k] = inB  + (uint64_t)src_row * ROW_BYTES + (uint64_t)chunk * 16u;
        dst_addr[k] = outB + (uint64_t)out_row * ROW_BYTES + (uint64_t)chunk * 16u;
        lds_off[k]  = ldsBase + c * 16u;
    }

    // Stage 16B/lane chunks into LDS via the CDNA5 async data mover (ASYNCcnt).
#pragma unroll
    for (int k = 0; k < CHUNKS_PER_THREAD; ++k) {
        asm volatile("global_load_async_to_lds_b128 %0, %1, off"
                     :: "v"(lds_off[k]), "v"(src_addr[k])
                     : "memory");
    }
    // Loads decrement ASYNCcnt only after the LDS write completes.
    asm volatile("s_wait_asynccnt 0x0" ::: "memory");
#pragma unroll
    for (int k = 0; k < CHUNKS_PER_THREAD; ++k) {
        asm volatile("global_store_async_from_lds_b128 %0, %1, off"
                     :: "v"(dst_addr[k]), "v"(lds_off[k])
                     : "memory");
    }
    // s_endpgm performs an implicit wait-idle, covering the outstanding stores.
}

extern "C" void kernel_launch(void* const* d_in, const int* in_sizes, int n_in,
                              void* d_out, int out_size, void* d_ws, size_t ws_size,
                              hipStream_t stream) {
    (void)in_sizes; (void)n_in; (void)out_size; (void)d_ws; (void)ws_size;
    const float* in = (const float*)d_in[0];
    float* out = (float*)d_out;

    const int blocks = ROWS_TOTAL / ROWS_PER_BLOCK;  // 1024
    cnot_gather_async<<<blocks, 256, 0, stream>>>(in, out);
}